// AggregPolicy_56839597195500
// MI455X (gfx1250) — compile-verified
//
#include <hip/hip_runtime.h>
#include <hip/hip_bf16.h>

// CDNA5 (gfx1250) wave32 implementation.
// Each wave handles 16 batch rows: gates on the WMMA M dimension, batch rows
// on N. All GEMMs use V_WMMA_F32_16X16X4_F32 (exact fp32; K=4 matches H=4).
// Gate-row mapping duplicates the z-gates into A rows 12..15 so BOTH lane
// halves compute z locally; only r (low->high) and n (high->low) cross the
// half-wave boundary. ALL 4-wide state is kept as float4 SSA values (never
// indexed), so nothing can be demoted to LDS/scratch.

typedef float v2f __attribute__((ext_vector_type(2)));
typedef float v8f __attribute__((ext_vector_type(8)));

__device__ __forceinline__ float ftanh(float x) {
#if __has_builtin(__builtin_amdgcn_tanhf)
  return __builtin_amdgcn_tanhf(x);            // v_tanh_f32
#else
  float e = __builtin_amdgcn_exp2f(-2.8853900817779268f * x); // 2^(-2x*log2e)
  return 2.0f * __builtin_amdgcn_rcpf(1.0f + e) - 1.0f;
#endif
}
__device__ __forceinline__ float fsigmoid(float x) {
  return fmaf(0.5f, ftanh(0.5f * x), 0.5f);
}

// Cross the half-wave boundary: lane i <-> lane i^16 (full-rate VALU).
__device__ __forceinline__ float swap16(float v) {
#if __has_builtin(__builtin_amdgcn_permlanex16)
  int i = __builtin_bit_cast(int, v);
  i = __builtin_amdgcn_permlanex16(i, i, 0x76543210, 0xfedcba98, false, false);
  return __builtin_bit_cast(float, i);
#else
  return __shfl_xor(v, 16, 32);
#endif
}

__device__ __forceinline__ v8f wmma4(v2f a, v2f b, v8f c) {
  // D[16x16] = A[16x4] * B[4x16] + C   (fp32, exact)
  return __builtin_amdgcn_wmma_f32_16x16x4_f32(false, a, false, b, (short)0, c,
                                               false, false);
}

// B fragment for a 4x16 fp32 B operand: VGPR0 = (lo? K0 : K2), VGPR1 = (lo? K1 : K3).
// Pure selects between SSA scalars -> v_cndmask, never memory.
__device__ __forceinline__ v2f bfrag(float4 h, bool hi) {
  v2f b;
  b[0] = hi ? h.z : h.x;
  b[1] = hi ? h.w : h.y;
  return b;
}

// One gate lane of the batched GRUCell update.
// (dn,en) = n/r pre-activations (low half: r rows; high half: n rows),
// (dz,ez) = z pre-activations (valid in BOTH halves via z-row duplication).
__device__ __forceinline__ float gate_k(float dn, float en, float dz, float ez,
                                        float hk, bool hi) {
  float t  = fsigmoid(dn + en);          // low: r_k
  float ts = swap16(t);                  // unconditional cross-lane
  float r  = hi ? ts : t;                // everyone: r_k
  float u  = ftanh(fmaf(r, en, dn));     // high: n_k = tanh(inn + r*hn)
  float us = swap16(u);
  float nn = hi ? u : us;                // everyone: n_k
  float z  = fsigmoid(dz + ez);          // both halves compute z locally
  return fmaf(z, hk - nn, nn);           // (1-z)*n + z*h
}

__device__ __forceinline__ float4 gru_update(v8f d, v8f e, float4 h, bool hi) {
  float4 o;
  o.x = gate_k(d[0], e[0], d[4], e[4], h.x, hi);
  o.y = gate_k(d[1], e[1], d[5], e[5], h.y, hi);
  o.z = gate_k(d[2], e[2], d[6], e[6], h.z, hi);
  o.w = gate_k(d[3], e[3], d[7], e[7], h.w, hi);
  return o;
}

__global__ __launch_bounds__(256, 1) void aggreg_policy_kernel(
    const float* __restrict__ x,
    const float* __restrict__ Wj,    const float* __restrict__ bj,
    const float* __restrict__ Wm,    const float* __restrict__ bm,
    const float* __restrict__ Wih_j, const float* __restrict__ Whh_j,
    const float* __restrict__ bih_j, const float* __restrict__ bhh_j,
    const float* __restrict__ Wih_m, const float* __restrict__ Whh_m,
    const float* __restrict__ bih_m, const float* __restrict__ bhh_m,
    const float* __restrict__ Wa,    const float* __restrict__ ba,
    float* __restrict__ out, int B) {
  const int lane = threadIdx.x & 31;
  const int wave = threadIdx.x >> 5;
  const bool hi  = (lane & 16) != 0;
  const int col  = lane & 15;                       // batch column within wave
  const int rowBase = (blockIdx.x * (blockDim.x >> 5) + wave) << 4;
  if (rowBase >= B) return;                         // wave-uniform exit
  int row = rowBase + col;
  if (row >= B) row = B - 1;                        // safety clamp

  // ---- per-row inputs (both lane halves hold the full row; 8B aligned) ----
  const float2* xr2 = reinterpret_cast<const float2*>(x + (size_t)row * 18);
  float xv[18];
#pragma unroll
  for (int i = 0; i < 9; ++i) {
    float2 p = xr2[i];
    xv[2 * i] = p.x;
    xv[2 * i + 1] = p.y;
  }

  // ---- A fragments: row m holds gate g = (m<12 ? m : m-8); half selects K pair
  const int m  = col;
  const int g_row = (m < 12) ? m : (m - 8);         // rows 12..15 = z-gate dup
  const int k0 = hi ? 2 : 0;
  v2f A_mih, A_mhh, A_jlo, A_jhi, A_jhh;
  A_mih[0] = Wih_m[g_row * 4 + k0];     A_mih[1] = Wih_m[g_row * 4 + k0 + 1];
  A_mhh[0] = Whh_m[g_row * 4 + k0];     A_mhh[1] = Whh_m[g_row * 4 + k0 + 1];
  A_jlo[0] = Wih_j[g_row * 8 + k0];     A_jlo[1] = Wih_j[g_row * 8 + k0 + 1];
  A_jhi[0] = Wih_j[g_row * 8 + 4 + k0]; A_jhi[1] = Wih_j[g_row * 8 + 4 + k0 + 1];
  A_jhh[0] = Whh_j[g_row * 4 + k0];     A_jhh[1] = Whh_j[g_row * 4 + k0 + 1];

  // ---- bias C fragments (D layout; same gate-row duplication) ----
  // bih and bhh stay separate: the n-gate is tanh(inn + r*(hn + bhh_n)).
  v8f Cbi_j, Cbh_j, Cbi_m, Cbh_m;
#pragma unroll
  for (int k = 0; k < 8; ++k) {
    int mm = hi ? (8 + k) : k;
    int g = (mm < 12) ? mm : (mm - 8);
    Cbi_j[k] = bih_j[g];
    Cbh_j[k] = bhh_j[g];
    Cbi_m[k] = bih_m[g];
    Cbh_m[k] = bhh_m[g];
  }

  // ---- encoders (trivial VALU; weights come in as s_loads) ----
  float4 hj[7], hm;
#pragma unroll
  for (int t = 0; t < 7; ++t) {
    float jv = xv[4 + t], jd = xv[11 + t];
    hj[t].x = fmaf(jv, Wj[0], fmaf(jd, Wj[1], bj[0]));
    hj[t].y = fmaf(jv, Wj[2], fmaf(jd, Wj[3], bj[1]));
    hj[t].z = fmaf(jv, Wj[4], fmaf(jd, Wj[5], bj[2]));
    hj[t].w = fmaf(jv, Wj[6], fmaf(jd, Wj[7], bj[3]));
  }
  hm.x = fmaf(xv[0], Wm[0],  fmaf(xv[1], Wm[1],  fmaf(xv[2], Wm[2],  fmaf(xv[3], Wm[3],  bm[0]))));
  hm.y = fmaf(xv[0], Wm[4],  fmaf(xv[1], Wm[5],  fmaf(xv[2], Wm[6],  fmaf(xv[3], Wm[7],  bm[1]))));
  hm.z = fmaf(xv[0], Wm[8],  fmaf(xv[1], Wm[9],  fmaf(xv[2], Wm[10], fmaf(xv[3], Wm[11], bm[2]))));
  hm.w = fmaf(xv[0], Wm[12], fmaf(xv[1], Wm[13], fmaf(xv[2], Wm[14], fmaf(xv[3], Wm[15], bm[3]))));

  // ---- 7 recurrence steps, fully unrolled: 22 WMMAs per step per wave ----
#pragma unroll
  for (int it = 0; it < 7; ++it) {
    // message GRU: x = hj[0] (old), h = hm (old)
    v8f dm = wmma4(A_mih, bfrag(hj[0], hi), Cbi_m);
    v8f em = wmma4(A_mhh, bfrag(hm, hi), Cbh_m);
    float4 hm_new = gru_update(dm, em, hm, hi);

    float4 prev = hm;                       // left operand for t = 0 (old hm)
#pragma unroll
    for (int t = 0; t < 7; ++t) {
      float4 cur = hj[t];                   // old hj[t]
      // gi = Wih_j[:,0:4]*left + Wih_j[:,4:8]*right + bih_j (chained through C)
      v8f d1 = wmma4(A_jlo, bfrag(prev, hi), Cbi_j);
      if (t != 6)                           // right neighbor of t=6 is zero
        d1 = wmma4(A_jhi, bfrag(hj[t + 1], hi), d1);
      v8f d2 = wmma4(A_jhh, bfrag(cur, hi), Cbh_j);
      hj[t] = gru_update(d1, d2, cur, hi);
      prev = cur;                           // old hj[t] = left for t+1
    }
    hm = hm_new;
  }

  // ---- head: acts[row][t] = hj[t] . Wa + ba  (low half stores) ----
  if (!hi && (rowBase + col) < B) {
    float* o = out + (size_t)row * 7;
#pragma unroll
    for (int t = 0; t < 7; ++t) {
      float a = fmaf(hj[t].x, Wa[0],
               fmaf(hj[t].y, Wa[1],
               fmaf(hj[t].z, Wa[2],
               fmaf(hj[t].w, Wa[3], ba[0]))));
      o[t] = a;
    }
  }
}

extern "C" void kernel_launch(void* const* d_in, const int* in_sizes, int n_in,
                              void* d_out, int out_size, void* d_ws, size_t ws_size,
                              hipStream_t stream) {
  (void)n_in; (void)d_ws; (void)ws_size; (void)out_size;
  const float* x     = (const float*)d_in[0];
  const float* Wj    = (const float*)d_in[1];
  const float* bj    = (const float*)d_in[2];
  const float* Wm    = (const float*)d_in[3];
  const float* bm    = (const float*)d_in[4];
  const float* Wih_j = (const float*)d_in[5];
  const float* Whh_j = (const float*)d_in[6];
  const float* bih_j = (const float*)d_in[7];
  const float* bhh_j = (const float*)d_in[8];
  const float* Wih_m = (const float*)d_in[9];
  const float* Whh_m = (const float*)d_in[10];
  const float* bih_m = (const float*)d_in[11];
  const float* bhh_m = (const float*)d_in[12];
  const float* Wa    = (const float*)d_in[13];
  const float* ba    = (const float*)d_in[14];

  const int B = in_sizes[0] / 18;
  const int rowsPerBlock = 128;  // 256 threads = 8 wave32 waves * 16 rows
  const int grid = (B + rowsPerBlock - 1) / rowsPerBlock;
  aggreg_policy_kernel<<<grid, 256, 0, stream>>>(
      x, Wj, bj, Wm, bm, Wih_j, Whh_j, bih_j, bhh_j,
      Wih_m, Whh_m, bih_m, bhh_m, Wa, ba, (float*)d_out, B);
}